// M1_5514738008540
// MI455X (gfx1250) — compile-verified
//
#include <hip/hip_runtime.h>

#define N_NODES 100000
#define N_EDGES 1600000
#define F_IN    32
#define HDIM    64
#define NEG_SLOPE 0.01f
#define BN_EPS    1e-5f

typedef __attribute__((ext_vector_type(2))) float v2f;
typedef __attribute__((ext_vector_type(8))) float v8f;

__device__ __forceinline__ float leaky_f(float v) { return v >= 0.f ? v : NEG_SLOPE * v; }

// ---------------- zero ----------------
__global__ void zero_kernel(float* __restrict__ p, int n) {
    int i = blockIdx.x * blockDim.x + threadIdx.x;
    if (i < n) p[i] = 0.f;
}

// ---------------- edge scatter-add (GIN aggregation) ----------------
// one thread per (edge, feature); d consecutive lanes share an edge -> coalesced
__global__ void scatter_kernel(const float* __restrict__ x, const int* __restrict__ src,
                               const int* __restrict__ dst, float* __restrict__ agg,
                               int nE, int dlog2) {
    long long i = (long long)blockIdx.x * blockDim.x + threadIdx.x;
    long long total = (long long)nE << dlog2;
    if (i >= total) return;
    int d = 1 << dlog2;
    int f = (int)(i & (d - 1));
    int e = (int)(i >> dlog2);
    float v = x[(long long)src[e] * d + f];
    atomicAdd(&agg[(long long)dst[e] * d + f], v);
}

// ---------------- WMMA GEMM: Out[N x 64] = A[N x K] @ W[K x 64] + bias ----------------
// MODE 0: A = (1+eps)*X + AGG           (GIN combine fused into A fetch)
// MODE 1: A = leaky(X*scale[k]+shift[k]) (BN + LeakyReLU fused into A fetch)
// Uses V_WMMA_F32_16X16X4_F32 (exact f32, matches reference numerics).
template <int MODE>
__global__ void gemm_wmma_kernel(const float* __restrict__ X, const float* __restrict__ AGG,
                                 const float* __restrict__ epsPtr,
                                 const float* __restrict__ W, const float* __restrict__ bias,
                                 const float* __restrict__ scale, const float* __restrict__ shift,
                                 float* __restrict__ Out, int K) {
    const int wave = blockIdx.x * (blockDim.x >> 5) + (threadIdx.x >> 5);
    const int lane = threadIdx.x & 31;
    const int mt = wave >> 2;           // HDIM/16 == 4 column tiles
    const int nt = wave & 3;
    if (mt * 16 >= N_NODES) return;     // wave-uniform; exact division in practice

    __builtin_prefetch(W, 0, 3);        // global_prefetch_b8 of weight tile

    const int mrow = mt * 16 + (lane & 15);        // A row for this lane
    const int ncol = nt * 16 + (lane & 15);        // B/C column for this lane
    const int koff = (lane >> 4) * 2;              // ISA A/B 16x4 f32 lane layout

    float e1 = 1.0f;
    if (MODE == 0) e1 = 1.0f + epsPtr[0];

    v8f c = {0.f, 0.f, 0.f, 0.f, 0.f, 0.f, 0.f, 0.f};

    for (int k0 = 0; k0 < K; k0 += 4) {
        const int ka = k0 + koff;
        const int kb = ka + 1;
        v2f a, b;
        if (MODE == 0) {
            const long long base = (long long)mrow * K;
            a.x = e1 * X[base + ka] + AGG[base + ka];
            a.y = e1 * X[base + kb] + AGG[base + kb];
        } else {
            const long long base = (long long)mrow * K;
            a.x = leaky_f(X[base + ka] * scale[ka] + shift[ka]);
            a.y = leaky_f(X[base + kb] * scale[kb] + shift[kb]);
        }
        b.x = W[ka * HDIM + ncol];
        b.y = W[kb * HDIM + ncol];
        c = __builtin_amdgcn_wmma_f32_16x16x4_f32(false, a, false, b, (short)0, c,
                                                  false, false);
    }

    const float bi = bias[ncol];
    const int rbase = mt * 16 + ((lane >> 4) ? 8 : 0);  // C layout: vgpr r -> M=r / M=8+r
#pragma unroll
    for (int r = 0; r < 8; ++r)
        Out[(long long)(rbase + r) * HDIM + ncol] = c[r] + bi;
}

// ---------------- BatchNorm partial sums (per-feature, over node dim) ----------------
__global__ void bn_partial_kernel(const float* __restrict__ hbuf, float* __restrict__ part,
                                  int Nrows) {
    __shared__ float ss[256];
    __shared__ float sq[256];
    const int f = threadIdx.x & 63;
    const int sub = threadIdx.x >> 6;   // 0..3
    const int base = blockIdx.x * 256;
    float s = 0.f, q = 0.f;
    for (int i = 0; i < 64; ++i) {
        int row = base + sub + i * 4;
        if (row < Nrows) {
            float v = hbuf[(long long)row * HDIM + f];
            s += v;
            q += v * v;
        }
    }
    ss[threadIdx.x] = s;
    sq[threadIdx.x] = q;
    __syncthreads();
    if (sub == 0) {
        float S = ss[f] + ss[f + 64] + ss[f + 128] + ss[f + 192];
        float Q = sq[f] + sq[f + 64] + sq[f + 128] + sq[f + 192];
        part[blockIdx.x * 128 + f] = S;
        part[blockIdx.x * 128 + 64 + f] = Q;
    }
}

// ---------------- BatchNorm finalize: scale/shift per feature ----------------
__global__ void bn_final_kernel(const float* __restrict__ part, int nblocks,
                                const float* __restrict__ gamma, const float* __restrict__ beta,
                                float* __restrict__ scaleshift) {
    int f = threadIdx.x;  // 64 threads
    float S = 0.f, Q = 0.f;
    for (int b = 0; b < nblocks; ++b) {
        S += part[b * 128 + f];
        Q += part[b * 128 + 64 + f];
    }
    const float invN = 1.0f / (float)N_NODES;
    float m = S * invN;
    float var = Q * invN - m * m;
    float rstd = rsqrtf(var + BN_EPS);
    float sc = gamma[f] * rstd;
    scaleshift[f] = sc;
    scaleshift[64 + f] = beta[f] - m * sc;
}

// ---------------- elementwise outer BN + LeakyReLU (in place) ----------------
__global__ void bnleaky_kernel(float* __restrict__ hbuf, const float* __restrict__ scaleshift,
                               int n) {
    int i = blockIdx.x * blockDim.x + threadIdx.x;
    if (i >= n) return;
    int f = i & 63;
    float v = hbuf[i] * scaleshift[f] + scaleshift[64 + f];
    hbuf[i] = leaky_f(v);
}

// ---------------- final projection: out[n] = x[n,:] . Wf + bf (one wave per node) ----------------
__global__ void final_dot_kernel(const float* __restrict__ X, const float* __restrict__ Wf,
                                 const float* __restrict__ bf, float* __restrict__ out, int Nn) {
    int wid = (blockIdx.x * blockDim.x + threadIdx.x) >> 5;
    int lane = threadIdx.x & 31;
    if (wid >= Nn) return;
    const long long base = (long long)wid * HDIM;
    float s = X[base + lane] * Wf[lane] + X[base + lane + 32] * Wf[lane + 32];
#pragma unroll
    for (int off = 16; off > 0; off >>= 1) s += __shfl_down(s, off, 32);
    if (lane == 0) out[wid] = s + bf[0];
}

extern "C" void kernel_launch(void* const* d_in, const int* in_sizes, int n_in,
                              void* d_out, int out_size, void* d_ws, size_t ws_size,
                              hipStream_t stream) {
    (void)in_sizes; (void)n_in; (void)out_size; (void)ws_size;

    const float* x = (const float*)d_in[0];
    const int* ei = (const int*)d_in[1];
    const int* src = ei;
    const int* dst = ei + N_EDGES;
    const float* W1[3] = {(const float*)d_in[2], (const float*)d_in[3], (const float*)d_in[4]};
    const float* b1[3] = {(const float*)d_in[5], (const float*)d_in[6], (const float*)d_in[7]};
    const float* g1[3] = {(const float*)d_in[8], (const float*)d_in[9], (const float*)d_in[10]};
    const float* be1[3] = {(const float*)d_in[11], (const float*)d_in[12], (const float*)d_in[13]};
    const float* W2[3] = {(const float*)d_in[14], (const float*)d_in[15], (const float*)d_in[16]};
    const float* b2[3] = {(const float*)d_in[17], (const float*)d_in[18], (const float*)d_in[19]};
    const float* eps[3] = {(const float*)d_in[20], (const float*)d_in[21], (const float*)d_in[22]};
    const float* go[3] = {(const float*)d_in[23], (const float*)d_in[24], (const float*)d_in[25]};
    const float* bo[3] = {(const float*)d_in[26], (const float*)d_in[27], (const float*)d_in[28]};
    const float* Wf = (const float*)d_in[29];
    const float* bf = (const float*)d_in[30];

    float* ws = (float*)d_ws;
    float* bufA = ws;                                   // GEMM2 out / x_cur  [N x 64]
    float* bufB = bufA + (size_t)N_NODES * HDIM;        // agg                [N x 64]
    float* bufC = bufB + (size_t)N_NODES * HDIM;        // GEMM1 out          [N x 64]
    float* part = bufC + (size_t)N_NODES * HDIM;        // BN partials
    const int NB_BN = (N_NODES + 255) / 256;            // 391
    float* scsh = part + (size_t)NB_BN * 128;           // scale[64] | shift[64]

    const dim3 blk(256);
    const int gemm_blocks = (N_NODES / 16) * (HDIM / 16) / 8;  // 3125

    for (int i = 0; i < 3; ++i) {
        const int K = (i == 0) ? F_IN : HDIM;
        const int dlog2 = (i == 0) ? 5 : 6;
        const float* xc = (i == 0) ? x : bufA;

        const int nagg = N_NODES * K;
        zero_kernel<<<(nagg + 255) / 256, blk, 0, stream>>>(bufB, nagg);

        const long long tot = (long long)N_EDGES << dlog2;
        scatter_kernel<<<(unsigned)((tot + 255) / 256), blk, 0, stream>>>(xc, src, dst, bufB,
                                                                          N_EDGES, dlog2);

        gemm_wmma_kernel<0><<<gemm_blocks, blk, 0, stream>>>(xc, bufB, eps[i], W1[i], b1[i],
                                                             nullptr, nullptr, bufC, K);

        bn_partial_kernel<<<NB_BN, blk, 0, stream>>>(bufC, part, N_NODES);
        bn_final_kernel<<<1, 64, 0, stream>>>(part, NB_BN, g1[i], be1[i], scsh);

        gemm_wmma_kernel<1><<<gemm_blocks, blk, 0, stream>>>(bufC, nullptr, nullptr, W2[i], b2[i],
                                                             scsh, scsh + 64, bufA, HDIM);

        if (i < 2) {
            bn_partial_kernel<<<NB_BN, blk, 0, stream>>>(bufA, part, N_NODES);
            bn_final_kernel<<<1, 64, 0, stream>>>(part, NB_BN, go[i], bo[i], scsh);
            bnleaky_kernel<<<(N_NODES * HDIM + 255) / 256, blk, 0, stream>>>(bufA, scsh,
                                                                             N_NODES * HDIM);
        }
    }

    final_dot_kernel<<<(N_NODES * 32 + 255) / 256, blk, 0, stream>>>(bufA, Wf, bf, (float*)d_out,
                                                                     N_NODES);
}